// BFGSCameraSolver_18992345382960
// MI455X (gfx1250) — compile-verified
//
#include <hip/hip_runtime.h>

// BFGS quadratic solver, one workgroup per (b,e) problem.
// H [128x128 f32] lives in LDS for the whole solve (row stride padded to 130
// floats for bank-conflict-free access). The rank-2 BFGS update is done with
// V_WMMA_F32_16X16X4_F32: the three rank-1 terms are packed into the K=4
// reduction slots, C/D = the 16x16 H tile (read-modify-write in LDS).

constexpr int N     = 128;
constexpr int LDH   = 130;   // padded row stride (floats) -> banks (2r+k)%64
constexpr int ITERS = 8;
constexpr float EPS2 = 1e-12f;  // (1e-6)^2

typedef float v2f __attribute__((ext_vector_type(2)));
typedef float v8f __attribute__((ext_vector_type(8)));

__device__ __forceinline__ float block_reduce_128(float v, float* red, int tid) {
    red[tid] = v;
    __syncthreads();
    #pragma unroll
    for (int off = 64; off > 0; off >>= 1) {
        if (tid < off) red[tid] += red[tid + off];
        __syncthreads();
    }
    float r = red[0];
    __syncthreads();
    return r;
}

__global__ __launch_bounds__(128)
void bfgs_solver_kernel(const float* __restrict__ Hinit,
                        const float* __restrict__ Q,
                        const float* __restrict__ bvec,
                        const float* __restrict__ x0,
                        float* __restrict__ xout)
{
    extern __shared__ float smem[];
    float* Hs  = smem;              // [128][130]
    float* xs  = Hs + N * LDH;      // x            [128]
    float* bs  = xs + N;            // b            [128]
    float* gv  = bs + N;            // gradient g   [128]
    float* dv  = gv + N;            // direction d  [128]
    float* sv  = dv + N;            // step s       [128]
    float* yv  = sv + N;            // dg (y)       [128]
    float* hy  = yv + N;            // H y          [128]
    float* xn  = hy + N;            // x_next       [128]
    float* red = xn + N;            // reduce pad   [128]

    const int tid  = threadIdx.x;        // 0..127 (one output row each)
    const int lane = tid & 31;
    const int wave = tid >> 5;           // 0..3
    const int be   = blockIdx.x;

    // ---- load state: H <- inv_hessian_init (identity), x, b ----
    for (int j = tid; j < N * N; j += 128) {
        int r = j >> 7, c = j & 127;
        Hs[r * LDH + c] = Hinit[j];
    }
    xs[tid] = x0[be * N + tid];
    bs[tid] = bvec[be * N + tid];
    __syncthreads();

    const float4* qrow = reinterpret_cast<const float4*>(Q) + tid * 32;

    // ---- g = Q x - b ----
    {
        float acc = 0.f;
        #pragma unroll 4
        for (int k4 = 0; k4 < 32; ++k4) {
            float4 q = qrow[k4];
            int k = k4 * 4;
            acc += q.x * xs[k] + q.y * xs[k + 1] + q.z * xs[k + 2] + q.w * xs[k + 3];
        }
        gv[tid] = acc - bs[tid];
    }
    __syncthreads();

    const int  m  = lane & 15;
    const bool lo = (lane < 16);

    bool updating = true;

    for (int it = 0; it < ITERS; ++it) {
        // ---- d = -H g  (LDS matvec, conflict-free with LDH=130) ----
        {
            float acc = 0.f;
            const float* hrow = Hs + tid * LDH;
            #pragma unroll 4
            for (int k = 0; k < N; ++k) acc += hrow[k] * gv[k];
            dv[tid] = -acc;
        }
        __syncthreads();

        // ---- exact line search: alpha = -(g.d) / max(d'Qd, 1e-12) ----
        float qd = 0.f;
        {
            #pragma unroll 4
            for (int k4 = 0; k4 < 32; ++k4) {
                float4 q = qrow[k4];
                int k = k4 * 4;
                qd += q.x * dv[k] + q.y * dv[k + 1] + q.z * dv[k + 2] + q.w * dv[k + 3];
            }
        }
        float dQd = block_reduce_128(dv[tid] * qd, red, tid);
        float gd  = block_reduce_128(gv[tid] * dv[tid], red, tid);
        float alpha = -gd / fmaxf(dQd, 1e-12f);

        float si  = alpha * dv[tid];
        float xni = xs[tid] + si;
        sv[tid] = si;
        xn[tid] = xni;
        __syncthreads();

        // ---- gn = Q x_next - b ;  y = gn - g ----
        float gni;
        {
            float acc = 0.f;
            #pragma unroll 4
            for (int k4 = 0; k4 < 32; ++k4) {
                float4 q = qrow[k4];
                int k = k4 * 4;
                acc += q.x * xn[k] + q.y * xn[k + 1] + q.z * xn[k + 2] + q.w * xn[k + 3];
            }
            gni = acc - bs[tid];
        }
        float yi = gni - gv[tid];
        yv[tid] = yi;
        __syncthreads();

        // ---- Hy = H y ----
        {
            float acc = 0.f;
            const float* hrow = Hs + tid * LDH;
            #pragma unroll 4
            for (int k = 0; k < N; ++k) acc += hrow[k] * yv[k];
            hy[tid] = acc;
        }
        float sdg  = block_reduce_128(si * yi, red, tid);      // s.y  (syncs cover hy[] too)
        float ihdg = block_reduce_128(yi * hy[tid], red, tid); // y'Hy
        float err2 = block_reduce_128(gni * gni, red, tid);    // |grad(x_next)|^2

        bool  nz   = (sdg != 0.f);
        float safe = nz ? sdg : 1.f;
        float c1   = nz ? (sdg + ihdg) / (safe * safe) : 0.f;
        float c2   = nz ? 1.f / safe : 0.f;

        // ---- H += c1*s s' - c2*(Hy s' + s (Hy)')  via V_WMMA_F32_16X16X4_F32
        // K-slot packing: A cols = [c1*s, -c2*Hy, -c2*s, 0], B rows = [s', s', Hy', 0]
        // Lane-half selection folded into scalar coefficients -> straight-line
        // fragment setup (no EXEC divergence around the WMMA).
        //   A 16x4: lanes 0-15 hold K=0,1 ; lanes 16-31 hold K=2,3
        //   B 4x16: vgpr0 -> rows K0/K1 ; vgpr1 -> rows K2/K3
        //   C/D 16x16: vgpr v -> row (ti*16 + v) lanes 0-15, +8 lanes 16-31
        const float ca = lo ? c1   : -c2;   // a.x coeff: K0=c1*s | K2=-c2*s
        const float cb = lo ? -c2  : 0.f;   // a.y coeff: K1=-c2*Hy | K3=0
        const float cd = lo ? 1.f  : 0.f;   // b.y mask:  K2=Hy'   | K3=0

        // Each wave owns two 16-row bands: ti = 2*wave + {0,1}; A hoisted per band.
        for (int ti2 = 0; ti2 < 2; ++ti2) {
            const int ti = wave * 2 + ti2;
            const int gi = ti * 16 + m;
            v2f a;
            a.x = ca * sv[gi];
            a.y = cb * hy[gi];
            const int rbase = ti * 16 + (lo ? 0 : 8);

            #pragma unroll
            for (int tj = 0; tj < 8; ++tj) {
                const int gj = tj * 16 + m;
                v2f b;
                b.x = sv[gj];
                b.y = cd * hy[gj];

                v8f c;
                #pragma unroll
                for (int v = 0; v < 8; ++v) c[v] = Hs[(rbase + v) * LDH + gj];

                c = __builtin_amdgcn_wmma_f32_16x16x4_f32(
                        /*neg_a=*/false, a, /*neg_b=*/false, b,
                        /*c_mod=*/(short)0, c, /*reuse_a=*/false, /*reuse_b=*/false);

                #pragma unroll
                for (int v = 0; v < 8; ++v) Hs[(rbase + v) * LDH + gj] = c[v];
            }
        }
        __syncthreads();

        // ---- masked advance (updating is block-uniform) ----
        if (updating) {
            xs[tid] = xni;
            gv[tid] = gni;
        }
        updating = updating && (err2 > EPS2);
        __syncthreads();
        if (!updating) break;   // frozen forever once converged -> output-equivalent
    }

    xout[be * N + tid] = xs[tid];
}

extern "C" void kernel_launch(void* const* d_in, const int* in_sizes, int n_in,
                              void* d_out, int out_size, void* d_ws, size_t ws_size,
                              hipStream_t stream) {
    const float* Hinit = (const float*)d_in[0];  // [n,n] identity
    const float* Q     = (const float*)d_in[1];  // [n,n]
    const float* bvec  = (const float*)d_in[2];  // [B,E,n]
    const float* x0    = (const float*)d_in[3];  // [B,E,n]
    float* xout = (float*)d_out;

    const int be_total = in_sizes[3] / N;        // B*E = 1024
    const size_t shmem = (size_t)(N * LDH + 9 * N) * sizeof(float);  // ~71 KB

    bfgs_solver_kernel<<<be_total, 128, shmem, stream>>>(Hinit, Q, bvec, x0, xout);
}